// SNN_p_18648747999739
// MI455X (gfx1250) — compile-verified
//
#include <hip/hip_runtime.h>
#include <hip/hip_bf16.h>

typedef __attribute__((ext_vector_type(2))) float v2f;
typedef __attribute__((ext_vector_type(8))) float v8f;

#define F_EDGE 128
#define F_OUT  64

// ---------------------------------------------------------------------------
// Kernel 1: X1p[E,64] = X1[E,128] @ W[128,64] + b  using V_WMMA_F32_16X16X4_F32
//
// Block = 256 threads = 8 waves; each wave computes one 16x64 output tile.
// W is pre-swizzled into LDS in B-fragment order [kstep][ntile][lane][2] so the
// inner loop reads conflict-free b64 fragments. A-fragments are loaded directly
// from global in the ISA 16x4 f32 A layout:
//   lane L<16 : M=L,   K = 4*ks + {0,1}
//   lane L>=16: M=L-16,K = 4*ks + {2,3}
// ---------------------------------------------------------------------------
__global__ void __launch_bounds__(256)
gemm_e2n_wmma(const float* __restrict__ X1,
              const float* __restrict__ W,
              const float* __restrict__ bvec,
              float* __restrict__ X1p, int E) {
    __shared__ float wfrag[32 * 4 * 32 * 2];   // 8192 floats = 32 KB

    const int tid = threadIdx.x;

    // Cooperative pre-swizzle of W into B-fragment layout.
    // flat index i = ((ks*4 + nt)*32 + lane)*2 + j
    for (int i = tid; i < 32 * 4 * 32 * 2; i += 256) {
        const int j    = i & 1;
        const int lane = (i >> 1) & 31;
        const int nt   = (i >> 6) & 3;
        const int ks   = i >> 8;
        const int k    = ks * 4 + ((lane >> 4) << 1) + j;  // 16x4 B layout (K)
        const int n    = nt * 16 + (lane & 15);            // (N)
        wfrag[i] = W[k * F_OUT + n];
    }
    __syncthreads();

    const int  lane = tid & 31;
    const int  wave = tid >> 5;
    const long tile = (long)blockIdx.x * 8 + wave;   // one 16-row tile per wave
    if (tile * 16 + 16 > (long)E) return;

    // Per-lane A pointer: row = tile*16 + (lane&15), col base = 2*(lane>=16)
    const float* arow =
        X1 + (tile * 16 + (lane & 15)) * F_EDGE + ((lane >> 4) << 1);

    v8f acc0 = {}, acc1 = {}, acc2 = {}, acc3 = {};

    for (int ks = 0; ks < 32; ++ks) {
        v2f a = *(const v2f*)(arow + ks * 4);
        const v2f* bf = (const v2f*)&wfrag[ks * 256];
        v2f b0 = bf[0 * 32 + lane];
        v2f b1 = bf[1 * 32 + lane];
        v2f b2 = bf[2 * 32 + lane];
        v2f b3 = bf[3 * 32 + lane];
        // (neg_a, A, neg_b, B, c_mod, C, reuse_a, reuse_b)
        acc0 = __builtin_amdgcn_wmma_f32_16x16x4_f32(false, a, false, b0,
                                                     (short)0, acc0, false, false);
        acc1 = __builtin_amdgcn_wmma_f32_16x16x4_f32(false, a, false, b1,
                                                     (short)0, acc1, false, false);
        acc2 = __builtin_amdgcn_wmma_f32_16x16x4_f32(false, a, false, b2,
                                                     (short)0, acc2, false, false);
        acc3 = __builtin_amdgcn_wmma_f32_16x16x4_f32(false, a, false, b3,
                                                     (short)0, acc3, false, false);
    }

    // Epilogue: bias + store. C/D layout: VGPR r -> M=r (lanes 0-15) or
    // M=r+8 (lanes 16-31); N = lane & 15 within each 16-col tile.
    const int   nlo  = lane & 15;
    const int   mhi  = (lane >> 4) * 8;
    const float bb0  = bvec[ 0 + nlo];
    const float bb1  = bvec[16 + nlo];
    const float bb2  = bvec[32 + nlo];
    const float bb3  = bvec[48 + nlo];
    float* out = X1p + tile * 16 * F_OUT;
#pragma unroll
    for (int r = 0; r < 8; ++r) {
        const int m = r + mhi;
        out[m * F_OUT +  0 + nlo] = acc0[r] + bb0;
        out[m * F_OUT + 16 + nlo] = acc1[r] + bb1;
        out[m * F_OUT + 32 + nlo] = acc2[r] + bb2;
        out[m * F_OUT + 48 + nlo] = acc3[r] + bb3;
    }
}

// ---------------------------------------------------------------------------
// Kernel 2: COO scatter  acc[row] += val * X1p[col]   (one nonzero per wave,
// each lane handles 2 consecutive floats -> coalesced 256B gather + f32 atomics
// into the L2-resident 25.6 MB accumulator).
// ---------------------------------------------------------------------------
__global__ void __launch_bounds__(256)
spmm_scatter(const int* __restrict__ rows, const int* __restrict__ cols,
             const float* __restrict__ vals, const float* __restrict__ X1p,
             float* __restrict__ acc, int nnz) {
    const int gwave = (int)((blockIdx.x * blockDim.x + threadIdx.x) >> 5);
    const int lane  = threadIdx.x & 31;
    if (gwave >= nnz) return;

    const int   r = rows[gwave];
    const int   c = cols[gwave];
    const float v = vals[gwave];

    const float* src = X1p + (long)c * F_OUT + lane * 2;
    float*       dst = acc + (long)r * F_OUT + lane * 2;

    v2f x = *(const v2f*)src;
    atomicAdd(dst + 0, v * x.x);
    atomicAdd(dst + 1, v * x.y);
}

// ---------------------------------------------------------------------------
// Kernel 3: PReLU (single shared slope) from workspace accumulator to d_out.
// ---------------------------------------------------------------------------
__global__ void __launch_bounds__(256)
prelu_kernel(const float* __restrict__ acc, const float* __restrict__ pw,
             float* __restrict__ out, int n) {
    const int i = blockIdx.x * blockDim.x + threadIdx.x;
    if (i >= n) return;
    const float a = pw[0];
    const float x = acc[i];
    out[i] = (x >= 0.0f) ? x : a * x;
}

extern "C" void kernel_launch(void* const* d_in, const int* in_sizes, int n_in,
                              void* d_out, int out_size, void* d_ws, size_t ws_size,
                              hipStream_t stream) {
    const float* X1      = (const float*)d_in[0];
    const int*   sp_rows = (const int*)d_in[1];
    const int*   sp_cols = (const int*)d_in[2];
    const float* sp_vals = (const float*)d_in[3];
    const float* W_e2n   = (const float*)d_in[4];
    const float* b_e2n   = (const float*)d_in[5];
    const float* prelu_w = (const float*)d_in[6];

    const int E   = in_sizes[0] / F_EDGE;   // 1,600,000
    const int nnz = in_sizes[1];            // 3,200,000
    const int nout = out_size;              // N_NODES * 64

    float* X1p    = (float*)d_ws;                   // E * 64 floats (410 MB)
    float* accbuf = X1p + (size_t)E * F_OUT;        // nout floats (25.6 MB)

    // Zero the aggregation buffer every call (deterministic across replays).
    hipMemsetAsync(accbuf, 0, (size_t)nout * sizeof(float), stream);

    // GEMM: one 16-row tile per wave, 8 waves per block.
    const int tiles   = (E + 15) / 16;
    const int gblocks = (tiles + 7) / 8;
    gemm_e2n_wmma<<<gblocks, 256, 0, stream>>>(X1, W_e2n, b_e2n, X1p, E);

    // Scatter: one nonzero per wave, 8 waves per block.
    const int sblocks = (nnz + 7) / 8;
    spmm_scatter<<<sblocks, 256, 0, stream>>>(sp_rows, sp_cols, sp_vals,
                                              X1p, accbuf, nnz);

    // PReLU into d_out.
    prelu_kernel<<<(nout + 255) / 256, 256, 0, stream>>>(accbuf, prelu_w,
                                                         (float*)d_out, nout);
}